// CTCBeamDecoder_7627861917728
// MI455X (gfx1250) — compile-verified
//
#include <hip/hip_runtime.h>
#include <hip/hip_bf16.h>
#include <stdint.h>

// CTC prefix beam search on gfx1250.
// One workgroup per batch element; entire scan state lives in LDS.
// The per-step hash-merge (eqf @ [Pb|Pnb]) is mapped to V_WMMA_F32_16X16X4_F32.

typedef float v2f __attribute__((ext_vector_type(2)));
typedef float v8f __attribute__((ext_vector_type(8)));

#define T_MAX   128
#define BATCH   8
#define SSYM    29
#define BLANKC  28
#define BEAM    8
#define CAND    240            // BEAM + BEAM*SSYM
#define CPAD    256            // padded to 16 M-tiles x 64 K-tiles
#define THRESH  1e-3f
#define HASH_MUL 1000003u

__global__ __launch_bounds__(256)
void ctc_beam_kernel(const float* __restrict__ x,      // [T, B, S]
                     const int*   __restrict__ lengths, // [B]
                     float*       __restrict__ out)     // [B*T + B + B] floats
{
    __shared__ float    pm[32];                       // pruned frame probs
    __shared__ uint32_t beamHash[BEAM];
    __shared__ float    beamPb[BEAM], beamPnb[BEAM];
    __shared__ int      beamLen[BEAM];
    __shared__ int      pref[2][BEAM][T_MAX];         // double-buffered prefixes
    __shared__ uint32_t cHash[CPAD];
    __shared__ float    cPbNb[CPAD * 2 + 32];         // interleaved {Pb,Pnb} per candidate
    __shared__ float    mPb[CPAD], mPnb[CPAD];        // merged sums (WMMA output)
    __shared__ float    score[CAND];
    __shared__ int      selIdx[BEAM];
    __shared__ int      nParent[BEAM], nExtC[BEAM], nWPos[BEAM], nLen[BEAM];
    __shared__ uint32_t nHash[BEAM];
    __shared__ float    nPb[BEAM], nPnb[BEAM];

    const int b    = blockIdx.x;
    const int tid  = threadIdx.x;
    const int lane = tid & 31;
    const int wave = tid >> 5;
    const int length = lengths[b];

    // ---- initial beam state (matches reference init) ----
    if (tid < BEAM) {
        beamHash[tid] = (tid == 0) ? 0u : (0xDEAD0000u + (uint32_t)tid);
        beamPb[tid]   = (tid == 0) ? 1.0f : 0.0f;
        beamPnb[tid]  = 0.0f;
        beamLen[tid]  = 0;
    }
    for (int e = tid; e < BEAM * T_MAX; e += 256)
        pref[0][e >> 7][e & 127] = -1;
    __syncthreads();

    int cur = 0;
    for (int t = 0; t < length; ++t) {
        // ---- phase 1: load + prune frame probabilities ----
        if (tid < 32) {
            float p = (tid < SSYM) ? x[((size_t)t * BATCH + b) * SSYM + tid] : 0.0f;
            pm[tid] = (p > THRESH) ? p : 0.0f;
        }
        if (tid == 0 && t + 1 < length)
            __builtin_prefetch(&x[((size_t)(t + 1) * BATCH + b) * SSYM], 0, 1);
        __syncthreads();

        // ---- phase 2: build 240 candidates (+16 inert pads) ----
        if (tid < CPAD) {
            uint32_t h; float cpb = 0.0f, cpnb = 0.0f;
            if (tid < BEAM) {                       // keep-same-prefix candidates
                int w = tid;
                int len = beamLen[w];
                int last = (len > 0) ? pref[cur][w][len - 1] : -1;
                float plast = (len > 0) ? pm[last < 0 ? 0 : last] : 0.0f;
                float tot = beamPb[w] + beamPnb[w];
                h    = beamHash[w];
                cpb  = pm[BLANKC] * tot;            // blank transition
                cpnb = plast * beamPnb[w];          // repeat last char
            } else if (tid < CAND) {                // extend-by-symbol candidates
                int w = (tid - BEAM) / SSYM;
                int s = (tid - BEAM) % SSYM;
                int len = beamLen[w];
                int last = (len > 0) ? pref[cur][w][len - 1] : -1;
                float tot = beamPb[w] + beamPnb[w];
                float base = (s == last) ? beamPb[w] : tot;
                h    = beamHash[w] * HASH_MUL + (uint32_t)(s + 1);
                cpnb = (s == BLANKC) ? 0.0f : pm[s] * base;
            } else {                                // pad rows: unique hash, zero prob
                h = 0xFFFF0000u + (uint32_t)tid;
            }
            cHash[tid]         = h;
            cPbNb[tid * 2 + 0] = cpb;
            cPbNb[tid * 2 + 1] = cpnb;
        }
        __syncthreads();

        // ---- phase 3: merged sums via V_WMMA_F32_16X16X4_F32 ----
        // [mPb|mPnb](256x2) = eq(256x256) @ [cPb|cPnb](256x2)
        // wave handles two 16-row M-tiles; 64 chained K-tiles of 4.
        {
            const int m0  = wave * 32;
            const int m1  = m0 + 16;
            const int row = lane & 15;
            const uint32_t h_i0 = cHash[m0 + row];
            const uint32_t h_i1 = cHash[m1 + row];
            const int koff = (lane >> 4) << 1;      // A/B: low lanes K0/K1, high lanes K2/K3
            const int col  = lane & 15;             // B/D column
            v8f acc0 = {}; v8f acc1 = {};
            for (int kt = 0; kt < 64; ++kt) {
                const int jA = kt * 4 + koff;
                const uint32_t hj0 = cHash[jA];
                const uint32_t hj1 = cHash[jA + 1];
                v2f a0, a1, bb;
                a0.x = (hj0 == h_i0) ? 1.0f : 0.0f;  // A tile rows of eq (exact 0/1 in f32)
                a0.y = (hj1 == h_i0) ? 1.0f : 0.0f;
                a1.x = (hj0 == h_i1) ? 1.0f : 0.0f;
                a1.y = (hj1 == h_i1) ? 1.0f : 0.0f;
                const float b0 = cPbNb[jA * 2 + (col & 1)];        // col0=Pb, col1=Pnb
                const float b1 = cPbNb[(jA + 1) * 2 + (col & 1)];
                bb.x = (col < 2) ? b0 : 0.0f;
                bb.y = (col < 2) ? b1 : 0.0f;
                acc0 = __builtin_amdgcn_wmma_f32_16x16x4_f32(
                           false, a0, false, bb, (short)0, acc0, false, false);
                acc1 = __builtin_amdgcn_wmma_f32_16x16x4_f32(
                           false, a1, false, bb, (short)0, acc1, false, false);
            }
            // D layout: col 0 lives in lanes 0/16 (rows r / r+8), col 1 in lanes 1/17.
            if (col < 2) {
                const int rbase = (lane >> 4) * 8;
                float* dst = (col == 0) ? mPb : mPnb;
#pragma unroll
                for (int r = 0; r < 8; ++r) {
                    dst[m0 + rbase + r] = acc0[r];
                    dst[m1 + rbase + r] = acc1[r];
                }
            }
        }
        __syncthreads();

        // ---- phase 4: canonical (first hash occurrence) + score ----
        if (tid < CAND) {
            const uint32_t h = cHash[tid];
            int fm = tid;
            for (int j = 0; j < tid; ++j) {
                if (cHash[j] == h) { fm = j; break; }
            }
            score[tid] = (fm == tid) ? (mPb[tid] + mPnb[tid]) : -1.0f;
        }
        __syncthreads();

        // ---- phase 5: stable top-8 (lax.top_k semantics) via exact ranking ----
        if (tid < CAND) {
            const float s = score[tid];
            int rank = 0;
            for (int j = 0; j < CAND; ++j) {
                const float sj = score[j];
                rank += (sj > s || (sj == s && j < tid)) ? 1 : 0;
            }
            if (rank < BEAM) selIdx[rank] = tid;
        }
        __syncthreads();

        // ---- phase 6: decode selections ----
        if (tid < BEAM) {
            const int w   = tid;
            const int idx = selIdx[w];
            const int parent = (idx < BEAM) ? idx : (idx - BEAM) / SSYM;
            const int extc   = (idx < BEAM) ? -1  : (idx - BEAM) % SSYM;
            nParent[w] = parent;
            nExtC[w]   = extc;
            nWPos[w]   = beamLen[parent];
            nLen[w]    = beamLen[parent] + (extc >= 0 ? 1 : 0);
            nHash[w]   = cHash[idx];
            nPb[w]     = mPb[idx];
            nPnb[w]    = mPnb[idx];
        }
        __syncthreads();

        // ---- phase 7: rebuild prefixes (double buffer) + commit scalars ----
        for (int e = tid; e < BEAM * T_MAX; e += 256) {
            const int w = e >> 7, pos = e & 127;
            int v = pref[cur][nParent[w]][pos];
            if (nExtC[w] >= 0 && pos == nWPos[w]) v = nExtC[w];
            pref[1 - cur][w][pos] = v;
        }
        if (tid < BEAM) {
            beamHash[tid] = nHash[tid];
            beamPb[tid]   = nPb[tid];
            beamPnb[tid]  = nPnb[tid];
            beamLen[tid]  = nLen[tid];
        }
        __syncthreads();
        cur = 1 - cur;
    }

    // ---- outputs: decoded[B,T], dec_lens[B], best_prob[B] (flattened as float) ----
    for (int pos = tid; pos < T_MAX; pos += 256)
        out[b * T_MAX + pos] = (float)pref[cur][0][pos];
    if (tid == 0) {
        out[BATCH * T_MAX + b]         = (float)beamLen[0];
        out[BATCH * T_MAX + BATCH + b] = beamPb[0] + beamPnb[0];
    }
}

extern "C" void kernel_launch(void* const* d_in, const int* in_sizes, int n_in,
                              void* d_out, int out_size, void* d_ws, size_t ws_size,
                              hipStream_t stream) {
    const float* x       = (const float*)d_in[0];   // [128, 8, 29] f32
    const int*   lengths = (const int*)d_in[1];     // [8] i32
    float*       out     = (float*)d_out;           // 8*128 + 8 + 8 = 1040 floats
    (void)in_sizes; (void)n_in; (void)out_size; (void)d_ws; (void)ws_size;
    ctc_beam_kernel<<<BATCH, 256, 0, stream>>>(x, lengths, out);
}